// DiffuSE_38620345926335
// MI455X (gfx1250) — compile-verified
//
#include <hip/hip_runtime.h>
#include <math.h>

// ---------------- model constants ----------------
static constexpr int B_ = 4, L_ = 2048, NS_ = 64, C_ = 64, E_ = 128;
static constexpr int DIN_ = 256, DSTATE_ = 64, DTRANK_ = 8, H_ = 512;
static constexpr int R_ = B_ * L_;                 // 8192 rows
static constexpr int NX_ = DTRANK_ + 2 * DSTATE_;  // 136
static constexpr int LDX_ = 144;                   // padded x_dbl stride
static constexpr float EPS_ = 1e-5f;
static constexpr float INV_SQRT_N_ = 0.022097086912079608f; // 1/sqrt(2048)

typedef float v2f __attribute__((ext_vector_type(2)));
typedef float v8f __attribute__((ext_vector_type(8)));

#define DEVFN static __device__ __forceinline__

DEVFN float sigmoidf_(float x) { return 1.0f / (1.0f + __expf(-x)); }
DEVFN float siluf_(float x) { return x * sigmoidf_(x); }
DEVFN float softplusf_(float x) { return (x > 20.f) ? x : log1pf(__expf(x)); }

// ---------------- WMMA f32 GEMM:  C[M x N] (+)= A[M x K] * W[N x K]^T ---------
// one wave -> 16x64 C strip (4 x 16x16 WMMA tiles), K stepped by 4.
// Tail-tile columns handled by CLAMPING the W row index (WMMA columns are
// independent; garbage columns never stored) -> branchless hot loop, EXEC
// stays all-ones. Two-stage software pipeline: k+4 fragments are loaded
// before the 4-wmma group of step k so loads stay in flight behind compute.
__global__ void k_gemm_wmma(const float* __restrict__ A, int lda,
                            const float* __restrict__ W,
                            float* __restrict__ Cm, int ldc,
                            int M, int N, int K, int accum) {
  int wave = (blockIdx.x * blockDim.x + threadIdx.x) >> 5;
  int lane = threadIdx.x & 31;
  int tnc = (N + 63) >> 6;
  int total = (M >> 4) * tnc;
  if (wave >= total) return;  // wave-uniform: EXEC stays all ones below
  int tm = wave / tnc, tn = wave % tnc;
  int row0 = tm << 4, col0 = tn << 6;
  int half = lane >> 4, lo = lane & 15;

  bool nok[4];
  int nidx[4];
#pragma unroll
  for (int t = 0; t < 4; ++t) {
    int n = col0 + t * 16 + lo;
    nok[t] = (n < N);
    nidx[t] = nok[t] ? n : (N - 1);  // clamp: memory-safe, result discarded
  }

  v8f acc[4];
#pragma unroll
  for (int t = 0; t < 4; ++t)
#pragma unroll
    for (int v = 0; v < 8; ++v) acc[t][v] = 0.f;
  if (accum) {
#pragma unroll
    for (int t = 0; t < 4; ++t)
#pragma unroll
      for (int v = 0; v < 8; ++v)
        acc[t][v] = Cm[(row0 + v + 8 * half) * (long)ldc + nidx[t]];
  }

  const float* ap = A + (long)(row0 + lo) * lda + 2 * half;
  const float* wp0 = W + (long)nidx[0] * K + 2 * half;
  const float* wp1 = W + (long)nidx[1] * K + 2 * half;
  const float* wp2 = W + (long)nidx[2] * K + 2 * half;
  const float* wp3 = W + (long)nidx[3] * K + 2 * half;

  // stage 0: preload k = 0
  v2f a, b0, b1, b2, b3;
  a.x = ap[0];    a.y = ap[1];
  b0.x = wp0[0];  b0.y = wp0[1];
  b1.x = wp1[0];  b1.y = wp1[1];
  b2.x = wp2[0];  b2.y = wp2[1];
  b3.x = wp3[0];  b3.y = wp3[1];

  for (int k0 = 4; k0 < K; k0 += 4) {
    // issue next-step loads first (distinct locals -> distinct regs, loads
    // overlap the wmma group below)
    v2f an, n0, n1, n2, n3;
    an.x = ap[k0];   an.y = ap[k0 + 1];
    n0.x = wp0[k0];  n0.y = wp0[k0 + 1];
    n1.x = wp1[k0];  n1.y = wp1[k0 + 1];
    n2.x = wp2[k0];  n2.y = wp2[k0 + 1];
    n3.x = wp3[k0];  n3.y = wp3[k0 + 1];

    acc[0] = __builtin_amdgcn_wmma_f32_16x16x4_f32(false, a, false, b0, (short)0, acc[0], false, false);
    acc[1] = __builtin_amdgcn_wmma_f32_16x16x4_f32(false, a, false, b1, (short)0, acc[1], false, false);
    acc[2] = __builtin_amdgcn_wmma_f32_16x16x4_f32(false, a, false, b2, (short)0, acc[2], false, false);
    acc[3] = __builtin_amdgcn_wmma_f32_16x16x4_f32(false, a, false, b3, (short)0, acc[3], false, false);

    a = an; b0 = n0; b1 = n1; b2 = n2; b3 = n3;
  }
  acc[0] = __builtin_amdgcn_wmma_f32_16x16x4_f32(false, a, false, b0, (short)0, acc[0], false, false);
  acc[1] = __builtin_amdgcn_wmma_f32_16x16x4_f32(false, a, false, b1, (short)0, acc[1], false, false);
  acc[2] = __builtin_amdgcn_wmma_f32_16x16x4_f32(false, a, false, b2, (short)0, acc[2], false, false);
  acc[3] = __builtin_amdgcn_wmma_f32_16x16x4_f32(false, a, false, b3, (short)0, acc[3], false, false);

#pragma unroll
  for (int t = 0; t < 4; ++t) {
    if (nok[t]) {
      int n = col0 + t * 16 + lo;
#pragma unroll
      for (int v = 0; v < 8; ++v)
        Cm[(row0 + v + 8 * half) * (long)ldc + n] = acc[t][v];
    }
  }
}

// ---------------- layernorm over last dim (=128), thread per row -------------
__global__ void k_layernorm128(const float* __restrict__ X, const float* __restrict__ g,
                               const float* __restrict__ bb, float* __restrict__ O) {
  int r = blockIdx.x * blockDim.x + threadIdx.x;
  if (r >= R_) return;
  const float* x = X + (long)r * E_;
  float m = 0.f;
  for (int e = 0; e < E_; ++e) m += x[e];
  m *= (1.0f / E_);
  float v = 0.f;
  for (int e = 0; e < E_; ++e) { float d = x[e] - m; v += d * d; }
  v *= (1.0f / E_);
  float rs = rsqrtf(v + EPS_);
  float* o = O + (long)r * E_;
  for (int e = 0; e < E_; ++e) o[e] = (x[e] - m) * rs * g[e] + bb[e];
}

// X[r,:] += layernorm(XO[r,:])
__global__ void k_addln128(const float* __restrict__ XO, const float* __restrict__ g,
                           const float* __restrict__ bb, float* __restrict__ X) {
  int r = blockIdx.x * blockDim.x + threadIdx.x;
  if (r >= R_) return;
  const float* x = XO + (long)r * E_;
  float m = 0.f;
  for (int e = 0; e < E_; ++e) m += x[e];
  m *= (1.0f / E_);
  float v = 0.f;
  for (int e = 0; e < E_; ++e) { float d = x[e] - m; v += d * d; }
  v *= (1.0f / E_);
  float rs = rsqrtf(v + EPS_);
  float* o = X + (long)r * E_;
  for (int e = 0; e < E_; ++e) o[e] += (x[e] - m) * rs * g[e] + bb[e];
}

// ---------------- depthwise causal conv (k=4) + silu -------------------------
__global__ void k_dwconv_silu(const float* __restrict__ XZ, const float* __restrict__ w,
                              const float* __restrict__ bias, float* __restrict__ XMC,
                              int seq_len) {
  int idx = blockIdx.x * blockDim.x + threadIdx.x;
  if (idx >= R_ * DIN_) return;
  int r = idx / DIN_, d = idx % DIN_;
  int pos = r % seq_len;
  float acc = bias[d];
#pragma unroll
  for (int k = 0; k < 4; ++k) {
    int t = pos - 3 + k;
    if (t >= 0) acc += w[d * 4 + k] * XZ[(long)(r - 3 + k) * (2 * DIN_) + d];
  }
  XMC[(long)r * DIN_ + d] = siluf_(acc);
}

// ---------------- delta = softplus(dt @ dt_proj_w^T + b) ---------------------
__global__ void k_delta(const float* __restrict__ XDBL, const float* __restrict__ dtw,
                        const float* __restrict__ dtb, float* __restrict__ DEL) {
  int idx = blockIdx.x * blockDim.x + threadIdx.x;
  if (idx >= R_ * DIN_) return;
  int r = idx / DIN_, d = idx % DIN_;
  float acc = dtb[d];
#pragma unroll
  for (int j = 0; j < DTRANK_; ++j) acc += XDBL[(long)r * LDX_ + j] * dtw[d * DTRANK_ + j];
  DEL[(long)r * DIN_ + d] = softplusf_(acc);
}

// ---------------- selective scan: block = one sequence, thread = channel d ---
// Reads DEL (delta), XMC (u), XDBL (B,C), XZ(z at col DIN_..), A_log, D.
// Writes gated output IN PLACE over DEL:   DEL[r,d] = (y + u*D[d]) * silu(z)
__global__ void k_scan(const float* __restrict__ XMC, const float* __restrict__ XDBL,
                       float* __restrict__ DEL, const float* __restrict__ A_log,
                       const float* __restrict__ Dv, const float* __restrict__ XZ,
                       int seq_len) {
  int seq = blockIdx.x;
  int d = threadIdx.x;  // 0..255
  __shared__ float sB[DSTATE_], sC[DSTATE_];

  float negA[DSTATE_];
#pragma unroll
  for (int n = 0; n < DSTATE_; ++n) negA[n] = -__expf(A_log[d * DSTATE_ + n]);
  float h[DSTATE_];
#pragma unroll
  for (int n = 0; n < DSTATE_; ++n) h[n] = 0.f;
  float Dd = Dv[d];

  int r0 = seq * seq_len;
  for (int t = 0; t < seq_len; ++t) {
    int r = r0 + t;
    if (d < 2 * DSTATE_) {
      float v = XDBL[(long)r * LDX_ + DTRANK_ + d];
      if (d < DSTATE_) sB[d] = v; else sC[d - DSTATE_] = v;
    }
    __syncthreads();
    float dt = DEL[(long)r * DIN_ + d];
    float u = XMC[(long)r * DIN_ + d];
    float du = dt * u;
    float y = 0.f;
#pragma unroll
    for (int n = 0; n < DSTATE_; ++n) {
      float dA = __expf(dt * negA[n]);
      h[n] = dA * h[n] + du * sB[n];
      y += h[n] * sC[n];
    }
    float z = XZ[(long)r * (2 * DIN_) + DIN_ + d];
    DEL[(long)r * DIN_ + d] = (y + u * Dd) * siluf_(z);
    __syncthreads();
  }
}

// ---------------- batchnorm helpers ------------------------------------------
__global__ void k_bnrelu512(float* __restrict__ Hb, const float* __restrict__ g,
                            const float* __restrict__ bb, const float* __restrict__ m,
                            const float* __restrict__ v) {
  long idx = (long)blockIdx.x * blockDim.x + threadIdx.x;
  if (idx >= (long)R_ * H_) return;
  int c = (int)(idx % H_);
  float x = Hb[idx];
  Hb[idx] = fmaxf((x - m[c]) * rsqrtf(v[c] + EPS_) * g[c] + bb[c], 0.f);
}

__global__ void k_bn_add128(const float* __restrict__ V, const float* __restrict__ g,
                            const float* __restrict__ bb, const float* __restrict__ m,
                            const float* __restrict__ vv, float* __restrict__ X) {
  long idx = (long)blockIdx.x * blockDim.x + threadIdx.x;
  if (idx >= (long)R_ * E_) return;
  int c = (int)(idx % E_);
  X[idx] += (V[idx] - m[c]) * rsqrtf(vv[c] + EPS_) * g[c] + bb[c];
}

// ---------------- FFT-based FFN mixing ---------------------------------------
DEVFN void fft_stages(float* re, float* im, int tid, float sign) {
  for (int i = tid; i < L_; i += 256) {
    int j = __brev((unsigned)i) >> 21;  // 11-bit reversal
    if (j > i) {
      float tr = re[i]; re[i] = re[j]; re[j] = tr;
      float ti = im[i]; im[i] = im[j]; im[j] = ti;
    }
  }
  __syncthreads();
  for (int len = 2; len <= L_; len <<= 1) {
    int half = len >> 1;
    float ab = sign * 6.283185307179586f / (float)len;
    for (int idx = tid; idx < L_ / 2; idx += 256) {
      int j = idx & (half - 1);
      int i0 = ((idx ^ j) << 1) + j;
      int i1 = i0 + half;
      float wi, wr;
      __sincosf(ab * (float)j, &wi, &wr);
      float vr = re[i1], vi = im[i1];
      float tr = vr * wr - vi * wi, ti = vr * wi + vi * wr;
      float ur = re[i0], ui = im[i0];
      re[i0] = ur + tr; im[i0] = ui + ti;
      re[i1] = ur - tr; im[i1] = ui - ti;
    }
    __syncthreads();
  }
}

// one block per (b,h) column of Hbuf (B,L,H); in-place FFT->diag/relu->iFFT
__global__ void k_fft_ffn(float* __restrict__ Hb, const float* __restrict__ rmat,
                          const float* __restrict__ imat, const float* __restrict__ rbv,
                          const float* __restrict__ ibv) {
  __shared__ float re[L_];
  __shared__ float im[L_];
  int b = blockIdx.x / H_, h = blockIdx.x % H_;
  int tid = threadIdx.x;
  for (int i = tid; i < L_; i += 256) {
    re[i] = Hb[((long)(b * L_ + i)) * H_ + h];
    im[i] = 0.f;
  }
  __syncthreads();
  fft_stages(re, im, tid, -1.0f);
  float rd = rmat[(long)h * H_ + h], idg = imat[(long)h * H_ + h];
  float rbb = rbv[h], ibb = ibv[h];
  for (int i = tid; i < L_; i += 256) {
    float xr = re[i] * INV_SQRT_N_, xi = im[i] * INV_SQRT_N_;
    float ar = fmaxf(xr * rd - xi * idg + rbb, 0.f);
    float ai = fmaxf(xi * rd + xr * idg + ibb, 0.f);
    re[i] = ar; im[i] = ai;
  }
  __syncthreads();
  fft_stages(re, im, tid, 1.0f);
  for (int i = tid; i < L_; i += 256)
    Hb[((long)(b * L_ + i)) * H_ + h] = re[i] * INV_SQRT_N_;
}

// ---------------- front-end kernels ------------------------------------------
__global__ void k_inp(const float* __restrict__ gt, const float* __restrict__ w,
                      const float* __restrict__ bias, float* __restrict__ X0) {
  long idx = (long)blockIdx.x * blockDim.x + threadIdx.x;
  if (idx >= (long)B_ * C_ * L_) return;
  int b = (int)(idx / (C_ * L_));
  int c = (int)((idx / L_) % C_);
  int l = (int)(idx % L_);
  X0[idx] = fmaxf(gt[(long)b * L_ + l] * w[c] + bias[c], 0.f);
}

__global__ void k_gather_emb(const float* __restrict__ table, const int* __restrict__ step,
                             float* __restrict__ out) {
  int idx = blockIdx.x * blockDim.x + threadIdx.x;
  if (idx >= B_ * 128) return;
  int b = idx >> 7, j = idx & 127;
  out[idx] = table[step[b] * 128 + j];
}

__global__ void k_mlp(const float* __restrict__ in, const float* __restrict__ W,
                      const float* __restrict__ bias, float* __restrict__ out,
                      int N, int K, int act) {
  int idx = blockIdx.x * blockDim.x + threadIdx.x;
  if (idx >= B_ * N) return;
  int b = idx / N, n = idx % N;
  float acc = bias[n];
  for (int k = 0; k < K; ++k) acc += in[(long)b * K + k] * W[(long)n * K + k];
  out[idx] = act ? siluf_(acc) : acc;
}

__global__ void k_up(const float* __restrict__ spec, const float* __restrict__ w,
                     const float* __restrict__ bias, float* __restrict__ SUP) {
  long idx = (long)blockIdx.x * blockDim.x + threadIdx.x;
  if (idx >= (long)B_ * NS_ * L_) return;
  int b = (int)(idx / (NS_ * L_));
  int i = (int)((idx / L_) % NS_);
  int j = (int)(idx % L_);
  float acc = bias[0];
#pragma unroll
  for (int ki = 0; ki < 3; ++ki) {
#pragma unroll
    for (int kj = 0; kj < 3; ++kj) {
      int ii = i - 1 + ki, jj = j - 1 + kj;
      if (ii >= 0 && ii < NS_ && jj >= 0 && jj < L_)
        acc += w[(2 - ki) * 3 + (2 - kj)] * spec[((long)b * NS_ + ii) * L_ + jj];
    }
  }
  SUP[idx] = (acc > 0.f) ? acc : 0.4f * acc;
}

__global__ void k_cproj(const float* __restrict__ SUP, const float* __restrict__ w,
                        const float* __restrict__ bias, float* __restrict__ COND) {
  long idx = (long)blockIdx.x * blockDim.x + threadIdx.x;
  if (idx >= (long)B_ * E_ * L_) return;
  int b = (int)(idx / (E_ * L_));
  int oc = (int)((idx / L_) % E_);
  int l = (int)(idx % L_);
  float acc = bias[oc];
  for (int ns = 0; ns < NS_; ++ns)
    acc += w[oc * NS_ + ns] * SUP[((long)b * NS_ + ns) * L_ + l];
  COND[idx] = acc;
}

__global__ void k_dil(const float* __restrict__ X0, const float* __restrict__ DP,
                      const float* __restrict__ COND, const float* __restrict__ w,
                      const float* __restrict__ bias, float* __restrict__ Y) {
  long idx = (long)blockIdx.x * blockDim.x + threadIdx.x;
  if (idx >= (long)B_ * E_ * L_) return;
  int b = (int)(idx / (E_ * L_));
  int oc = (int)((idx / L_) % E_);
  int l = (int)(idx % L_);
  float acc = bias[oc] + COND[idx];
  for (int ic = 0; ic < C_; ++ic) {
    const float* x = X0 + ((long)b * C_ + ic) * L_;
    float dp = DP[b * C_ + ic];
#pragma unroll
    for (int k = 0; k < 3; ++k) {
      int pos = l - 1 + k;
      if (pos >= 0 && pos < L_)
        acc += w[(oc * C_ + ic) * 3 + k] * (x[pos] + dp);
    }
  }
  Y[idx] = acc;
}

// (B, E, L) -> (B, L, E)
__global__ void k_tr(const float* __restrict__ Y, float* __restrict__ X) {
  long idx = (long)blockIdx.x * blockDim.x + threadIdx.x;
  if (idx >= (long)R_ * E_) return;
  int r = (int)(idx / E_);
  int e = (int)(idx % E_);
  int b = r / L_, l = r % L_;
  X[idx] = Y[((long)b * E_ + e) * L_ + l];
}

// skip(1x1 conv,relu) + out(1x1 conv), fused; thread per (b,l)
__global__ void k_final(const float* __restrict__ X, const float* __restrict__ skw,
                        const float* __restrict__ skb, const float* __restrict__ ow,
                        const float* __restrict__ ob, float* __restrict__ out) {
  int idx = blockIdx.x * blockDim.x + threadIdx.x;
  if (idx >= R_) return;
  const float* xr = X + (long)idx * E_;
  float acc = ob[0];
  for (int c = 0; c < C_; ++c) {
    float s = skb[c];
    const float* w = skw + (long)c * E_;
    for (int e = 0; e < E_; ++e) s += w[e] * xr[e];
    acc += ow[c] * fmaxf(s, 0.f);
  }
  out[idx] = acc;
}

// =============================================================================
extern "C" void kernel_launch(void* const* d_in, const int* in_sizes, int n_in,
                              void* d_out, int out_size, void* d_ws, size_t ws_size,
                              hipStream_t stream) {
  (void)in_sizes; (void)n_in; (void)out_size; (void)ws_size;
  auto F = [&](int i) { return (const float*)d_in[i]; };

  const float* gt = F(0);
  const float* spec = F(1);
  const int* dstep = (const int*)d_in[2];
  // params in setup_inputs() insertion order
  const float *inp_w = F(3), *inp_b = F(4), *emb = F(5);
  const float *p1w = F(6), *p1b = F(7), *p2w = F(8), *p2b = F(9);
  const float *upw = F(10), *upb = F(11);
  const float *dilw = F(12), *dilb = F(13);
  const float *dpw = F(14), *dpb = F(15);
  const float *cpw = F(16), *cpb = F(17);
  // blocks: base 18, 29 tensors each; tail: 76..79

  float* ws = (float*)d_ws;
  float* X = ws;                        // R_*128
  float* XO = X + (long)R_ * E_;        // R_*128
  float* LN = XO + (long)R_ * E_;       // R_*128
  float* XZ = LN + (long)R_ * E_;       // R_*512  (also FFN H buffer)
  float* XMC = XZ + (long)R_ * 512;     // R_*256
  float* XDBL = XMC + (long)R_ * DIN_;  // R_*144
  float* DEL = XDBL + (long)R_ * LDX_;  // R_*256 (delta, then gated scan out)
  // front-end aliases inside XZ / XMC regions (used before block loop)
  float* X0 = XZ;
  float* SUP = XZ + (long)B_ * C_ * L_;
  float* COND = XZ + 2L * B_ * C_ * L_;
  float* Y2 = XZ + 4L * B_ * C_ * L_;
  float* DEMB0 = XMC;
  float* D1 = XMC + 1024;
  float* D2 = D1 + 2048;
  float* DP = D2 + 2048;

  auto gemm = [&](const float* A, int lda, const float* W, float* Cp, int ldc,
                  int M, int N, int K, int acc) {
    int tiles = (M / 16) * ((N + 63) / 64);
    int blocks = (tiles + 7) / 8;  // 8 waves (256 thr) per block
    k_gemm_wmma<<<blocks, 256, 0, stream>>>(A, lda, W, Cp, ldc, M, N, K, acc);
  };
#define LAUNCH(kern, n, ...) kern<<<((int)(((long)(n) + 255) / 256)), 256, 0, stream>>>(__VA_ARGS__)

  // ---------------- front end ----------------
  LAUNCH(k_inp, (long)B_ * C_ * L_, gt, inp_w, inp_b, X0);
  LAUNCH(k_gather_emb, B_ * 128, emb, dstep, DEMB0);
  LAUNCH(k_mlp, B_ * 512, DEMB0, p1w, p1b, D1, 512, 128, 1);
  LAUNCH(k_mlp, B_ * 512, D1, p2w, p2b, D2, 512, 512, 1);
  LAUNCH(k_mlp, B_ * 64, D2, dpw, dpb, DP, 64, 512, 0);
  LAUNCH(k_up, (long)B_ * NS_ * L_, spec, upw, upb, SUP);
  LAUNCH(k_cproj, (long)B_ * E_ * L_, SUP, cpw, cpb, COND);
  LAUNCH(k_dil, (long)B_ * E_ * L_, X0, DP, COND, dilw, dilb, Y2);
  LAUNCH(k_tr, (long)R_ * E_, Y2, X);

  // ---------------- transformer blocks ----------------
  for (int blk = 0; blk < 2; ++blk) {
    int pb = 18 + 29 * blk;
    const float *ln_g = F(pb + 0), *ln_b = F(pb + 1), *ipw = F(pb + 2);
    const float *cw = F(pb + 3), *cb = F(pb + 4), *xpw = F(pb + 5);
    const float *dtw = F(pb + 6), *dtb = F(pb + 7), *Alog = F(pb + 8);
    const float *Dv = F(pb + 9), *opw = F(pb + 10);
    const float *n1g = F(pb + 11), *n1b = F(pb + 12), *n2g = F(pb + 13), *n2b = F(pb + 14);
    const float *fc1 = F(pb + 15), *bn1g = F(pb + 16), *bn1b = F(pb + 17);
    const float *bn1m = F(pb + 18), *bn1v = F(pb + 19);
    const float *rmat = F(pb + 20), *imat = F(pb + 21), *rbv = F(pb + 22), *ibv = F(pb + 23);
    const float *fc2 = F(pb + 24), *bn2g = F(pb + 25), *bn2b = F(pb + 26);
    const float *bn2m = F(pb + 27), *bn2v = F(pb + 28);

    hipMemsetAsync(XO, 0, (size_t)R_ * E_ * sizeof(float), stream);

    // layernorm + in_proj identical for all splits -> hoisted
    LAUNCH(k_layernorm128, R_, X, ln_g, ln_b, LN);
    gemm(LN, E_, ipw, XZ, 512, R_, 512, E_, 0);  // xz = ln(x) @ in_proj^T

    const int splits[4] = {1, 2, 4, 8};
    for (int si = 0; si < 4; ++si) {
      int s = splits[si];
      int seq_len = L_ / s;
      int nseq = B_ * s;
      LAUNCH(k_dwconv_silu, (long)R_ * DIN_, XZ, cw, cb, XMC, seq_len);
      gemm(XMC, DIN_, xpw, XDBL, LDX_, R_, NX_, DIN_, 0);       // x_dbl
      LAUNCH(k_delta, (long)R_ * DIN_, XDBL, dtw, dtb, DEL);    // softplus(dt_proj)
      k_scan<<<nseq, 256, 0, stream>>>(XMC, XDBL, DEL, Alog, Dv, XZ, seq_len);
      gemm(DEL, DIN_, opw, XO, E_, R_, E_, DIN_, 1);            // XO += g @ out_proj^T
    }
    LAUNCH(k_addln128, R_, XO, n1g, n1b, X);  // x += ln(x_o)

    // FFN
    LAUNCH(k_layernorm128, R_, X, n2g, n2b, LN);
    gemm(LN, E_, fc1, XZ, H_, R_, H_, E_, 0);  // h = ln2 @ fc1^T  (XZ as H buf)
    LAUNCH(k_bnrelu512, (long)R_ * H_, XZ, bn1g, bn1b, bn1m, bn1v);
    k_fft_ffn<<<B_ * H_, 256, 0, stream>>>(XZ, rmat, imat, rbv, ibv);
    gemm(XZ, H_, fc2, LN, E_, R_, E_, H_, 0);  // out = h2 @ fc2^T
    LAUNCH(k_bn_add128, (long)R_ * E_, LN, bn2g, bn2b, bn2m, bn2v, X);
  }

  // ---------------- tail: skip conv + out conv ----------------
  LAUNCH(k_final, R_, X, F(76), F(77), F(78), F(79), (float*)d_out);
#undef LAUNCH
}